// SteinDiscriminator_41781441855509
// MI455X (gfx1250) — compile-verified
//
#include <hip/hip_runtime.h>
#include <hip/hip_bf16.h>
#include <math.h>

typedef __attribute__((ext_vector_type(2))) float v2f;
typedef __attribute__((ext_vector_type(8))) float v8f;

constexpr int Lc = 32, Tc = 64, Dc = 8192, Hc = 16;
constexpr int ROWS   = Lc * Tc;        // 2048 (l,t) rows of w
constexpr int TILE_M = 16;
constexpr int NTILES = ROWS / TILE_M;  // 128
constexpr int KSPLIT = 16;             // K partitions per tile
constexpr int KCHUNK = Dc / KSPLIT;    // 512
constexpr int TPW    = 2;              // row-tiles per wave (amortize B fragments)

// Workspace layout (in floats):
constexpr size_t WS_H    = 0;                        // h [L*H] = 512
constexpr size_t WS_S    = 512;                      // s = 1-h^2 [L*H]
constexpr size_t WS_TSUM = 1024;                     // per-tile trace sums [128]
constexpr size_t WS_W2T  = 2048;                     // W2 transposed [H][D] = 131072
constexpr size_t WS_PART = WS_W2T + (size_t)Hc * Dc; // per-(tile,kc): 256 A + 256 B
constexpr size_t WS_PART_SZ = (size_t)NTILES * KSPLIT * 512;  // 4 MB

// ---------------- h = tanh(u @ W1^T + b1), s = 1 - h^2 ----------------
__global__ void k_hidden(const float* __restrict__ u, const float* __restrict__ W1,
                         const float* __restrict__ b1, float* __restrict__ ws) {
    __shared__ float red[256];
    const int l  = blockIdx.x >> 4;
    const int hh = blockIdx.x & 15;
    const float* ur = u  + (size_t)l  * Dc;
    const float* wr = W1 + (size_t)hh * Dc;
    float p = 0.f;
    for (int k = threadIdx.x; k < Dc; k += 256) p = fmaf(ur[k], wr[k], p);
    red[threadIdx.x] = p;
    __syncthreads();
    for (int s = 128; s > 0; s >>= 1) {
        if (threadIdx.x < s) red[threadIdx.x] += red[threadIdx.x + s];
        __syncthreads();
    }
    if (threadIdx.x == 0) {
        float hv = tanhf(red[0] + b1[hh]);
        ws[WS_H + blockIdx.x] = hv;
        ws[WS_S + blockIdx.x] = 1.f - hv * hv;
    }
}

// ---------------- W2T[h][d] = W2[d][h]  (512 KB, one-time) ----------------
__global__ void k_w2t(const float* __restrict__ W2, float* __restrict__ ws) {
    const int idx = blockIdx.x * 256 + threadIdx.x;   // < H*D
    const int hh = idx >> 13;                         // D = 8192
    const int d  = idx & (Dc - 1);
    ws[WS_W2T + idx] = W2[(size_t)d * Hc + hh];
}

// ---------------- f = h @ W2^T + b2  (writes d_out[0 .. L*D)) ----------------
__global__ void k_forward(const float* __restrict__ W2, const float* __restrict__ b2,
                          const float* __restrict__ ws, float* __restrict__ fout) {
    const int idx = blockIdx.x * 256 + threadIdx.x;     // < L*D
    const int l = idx >> 13;
    const int d = idx & (Dc - 1);
    const float* hr  = ws + WS_H + (size_t)l * Hc;
    const float* w2r = W2 + (size_t)d * Hc;
    float acc = b2[d];
#pragma unroll
    for (int i = 0; i < Hc; ++i) acc = fmaf(hr[i], w2r[i], acc);
    fout[idx] = acc;
}

// -------- Hot path: A = w@W2 and B = w@W1^T via V_WMMA_F32_16X16X4_F32 --------
// One wave per (2 row-tiles x K-chunk of 512). Four interleaved WMMA chains;
// B fragments (W1 row-pair + W2T row-pair, both contiguous b64 loads) are
// shared across the two row-tiles, so loads:WMMA = 1:1 and the redundant
// L2-side B traffic equals the unique 64 MB HBM stream of w.
__global__ void __launch_bounds__(32) k_gemms(const float* __restrict__ w,
                                              const float* __restrict__ W1,
                                              float* __restrict__ ws) {
    const int pairIdx = blockIdx.x / KSPLIT;   // 0..63
    const int kc      = blockIdx.x % KSPLIT;
    const int lane = threadIdx.x;
    const int n    = lane & 15;   // A: row within tile; B: N column
    const int g    = lane >> 4;   // selects K pair {2g, 2g+1} within each K=4 step
    const int tile0 = pairIdx * TPW;
    const int k0    = kc * KCHUNK;

    // A fragments (16x4 f32): lane holds w[row][k + 2g .. +1] -> b64 loads
    const float* ap0 = w + (size_t)(tile0 * TILE_M + n) * Dc + k0 + 2 * g;
    const float* ap1 = ap0 + (size_t)TILE_M * Dc;
    // B fragment for w@W1^T: B[k][n] = W1[n][k] -> contiguous pair, b64 load
    const float* b1p = W1 + (size_t)n * Dc + k0 + 2 * g;
    // B fragment for w@W2:   B[k][n] = W2T[n][k] -> contiguous pair, b64 load
    const float* b2p = ws + WS_W2T + (size_t)n * Dc + k0 + 2 * g;

    v8f accA0 = {}, accB0 = {};   // tile0:  w@W2 , w@W1^T
    v8f accA1 = {}, accB1 = {};   // tile0+1
#pragma unroll 4
    for (int k = 0; k < KCHUNK; k += 4) {
        v2f a0 = *(const v2f*)(ap0 + k);
        v2f a1 = *(const v2f*)(ap1 + k);
        v2f bw = *(const v2f*)(b2p + k);
        v2f bb = *(const v2f*)(b1p + k);
        accA0 = __builtin_amdgcn_wmma_f32_16x16x4_f32(false, a0, false, bw,
                                                      (short)0, accA0, false, false);
        accB0 = __builtin_amdgcn_wmma_f32_16x16x4_f32(false, a0, false, bb,
                                                      (short)0, accB0, false, false);
        accA1 = __builtin_amdgcn_wmma_f32_16x16x4_f32(false, a1, false, bw,
                                                      (short)0, accA1, false, false);
        accB1 = __builtin_amdgcn_wmma_f32_16x16x4_f32(false, a1, false, bb,
                                                      (short)0, accB1, false, false);
    }

    // C/D layout: VGPR r, lane-half g, col n -> element (M = r + 8g, N = n)
    float* out0 = ws + WS_PART + (size_t)((tile0 + 0) * KSPLIT + kc) * 512;
    float* out1 = ws + WS_PART + (size_t)((tile0 + 1) * KSPLIT + kc) * 512;
#pragma unroll
    for (int r = 0; r < 8; ++r) {
        const int mi = r + 8 * g;
        out0[mi * 16 + n]       = accA0[r];
        out0[256 + mi * 16 + n] = accB0[r];
        out1[mi * 16 + n]       = accA1[r];
        out1[256 + mi * 16 + n] = accB1[r];
    }
}

// ---- Combine K-partials, apply s, reduce 16x16 tile -> one scalar per tile ----
__global__ void k_trace_tile(float* __restrict__ ws) {
    __shared__ float red[256];
    const int tile = blockIdx.x;
    const int tid  = threadIdx.x;           // element (m*16 + h) of the tile
    const float* base = ws + WS_PART + (size_t)(tile * KSPLIT) * 512;
    float sa = 0.f, sb = 0.f;
#pragma unroll
    for (int kc = 0; kc < KSPLIT; ++kc) {   // fixed order -> deterministic
        sa += base[(size_t)kc * 512 + tid];
        sb += base[(size_t)kc * 512 + 256 + tid];
    }
    const int l  = tile >> 2;               // 64/16 = 4 tiles per l
    const float sv = ws[WS_S + (size_t)l * Hc + (tid & 15)];
    red[tid] = sv * sa * sb;
    __syncthreads();
    for (int s = 128; s > 0; s >>= 1) {
        if (tid < s) red[tid] += red[tid + s];
        __syncthreads();
    }
    if (tid == 0) ws[WS_TSUM + tile] = red[0];
}

// ---- Final: sum 128 tile scalars in fixed order, scale by 1/(L*T) ----
__global__ void k_final(const float* __restrict__ ws, float* __restrict__ outscalar) {
    if (threadIdx.x == 0) {
        float s = 0.f;
        for (int i = 0; i < NTILES; ++i) s += ws[WS_TSUM + i];
        *outscalar = s * (1.0f / (float)(Lc * Tc));
    }
}

extern "C" void kernel_launch(void* const* d_in, const int* in_sizes, int n_in,
                              void* d_out, int out_size, void* d_ws, size_t ws_size,
                              hipStream_t stream) {
    const float* u  = (const float*)d_in[0];
    const float* w  = (const float*)d_in[1];
    const float* W1 = (const float*)d_in[2];
    const float* b1 = (const float*)d_in[3];
    const float* W2 = (const float*)d_in[4];
    const float* b2 = (const float*)d_in[5];
    float* out = (float*)d_out;
    float* ws  = (float*)d_ws;

    k_w2t<<<(Hc * Dc) / 256, 256, 0, stream>>>(W2, ws);
    k_hidden<<<Lc * Hc, 256, 0, stream>>>(u, W1, b1, ws);
    k_gemms<<<(NTILES / TPW) * KSPLIT, 32, 0, stream>>>(w, W1, ws);
    k_forward<<<(Lc * Dc) / 256, 256, 0, stream>>>(W2, b2, ws, out);
    k_trace_tile<<<NTILES, 256, 0, stream>>>(ws);
    k_final<<<1, 32, 0, stream>>>(ws, out + (size_t)Lc * Dc);
}